// ModelNew_17411797418162
// MI455X (gfx1250) — compile-verified
//
#include <hip/hip_runtime.h>
#include <hip/hip_bf16.h>

// Fused residual-stream mixer for MI455X (gfx1250, wave32).
//   mixed = h_res @ x   (per (b,s): [4x4] @ [4x1024], K=4)
//   out   = mixed * h_out + h_post * x
// Uses V_WMMA_F32_16X16X4_F32 (exact f32 path, K=4 == stream dim).
// A-matrix rows: streams {0,1} -> M={0,1}, streams {2,3} -> M={8,9} so that
// C rows for streams 2,3 land in lanes 16-31 (C layout: VGPR r holds M=r in
// lanes 0-15 and M=r+8 in lanes 16-31) -- no cross-lane shuffles needed.

typedef __attribute__((ext_vector_type(2))) float v2f;
typedef __attribute__((ext_vector_type(8))) float v8f;

#define B_DIM 8
#define S_DIM 2048
#define N_DIM 4
#define D_DIM 1024

__global__ __launch_bounds__(256) void stream_mix_wmma(
    const float* __restrict__ x,      // [B,S,N,D]
    const float* __restrict__ h_res,  // [B,S,N,N]
    const float* __restrict__ h_out,  // [B,S,D]
    const float* __restrict__ h_post, // [B,S,N]
    float* __restrict__ out)          // [B,S,N,D]
{
    const int lane      = threadIdx.x & 31;
    const int waveInBlk = threadIdx.x >> 5;
    const int gw        = blockIdx.x * 8 + waveInBlk; // global wave id
    const int bs        = gw >> 2;                    // (b,s) flat index
    const int d_base    = (gw & 3) << 8;              // 256-wide D quarter

    const int n  = lane & 15;   // column within 16-wide tile / A-matrix row M
    const int h  = lane >> 4;   // 0 = lower half-wave, 1 = upper half-wave
    const int j0 = h * 2;       // x streams this lane handles: j0, j0+1

    // ---- Build A (16x4 f32): lane m holds row M=m; VGPR0/1 = K pair (2h, 2h+1).
    //      Rows 0,1 <- h_res streams 0,1 ; rows 8,9 <- streams 2,3 ; rest zero.
    const float* hr = h_res + (long)bs * (N_DIM * N_DIM);
    const int  m     = n;
    const bool vrow  = (m < 2) || (m == 8) || (m == 9);
    const int  si    = (m & 1) + ((m >> 3) << 1);     // stream index 0..3
    const int  kk    = h * 2;                         // K pair for this half
    v2f a;
    a.x = vrow ? hr[si * 4 + kk]     : 0.0f;
    a.y = vrow ? hr[si * 4 + kk + 1] : 0.0f;

    // ---- Per-stream residual scales for this lane's two streams.
    const float4 hp4 = *(const float4*)(h_post + (long)bs * N_DIM);
    const float hp0 = (h == 0) ? hp4.x : hp4.z;       // stream j0
    const float hp1 = (h == 0) ? hp4.y : hp4.w;       // stream j0+1

    const float* xb = x     + (long)bs * (N_DIM * D_DIM);
    float*       ob = out   + (long)bs * (N_DIM * D_DIM);
    const float* ho = h_out + (long)bs * D_DIM;

    #pragma unroll 4
    for (int c = 0; c < 16; ++c) {
        const int d0 = d_base + c * 16;

        // ---- B (4x16 f32): lane = column n; lower half rows K={0,1},
        //      upper half rows K={2,3}. Streaming loads (single use).
        v2f b;
        b.x = __builtin_nontemporal_load(&xb[(j0    ) * D_DIM + d0 + n]);
        b.y = __builtin_nontemporal_load(&xb[(j0 + 1) * D_DIM + d0 + n]);

        v8f acc = {};
        // (neg_a, A, neg_b, B, c_mod, C, reuse_a, reuse_b)
        acc = __builtin_amdgcn_wmma_f32_16x16x4_f32(
            false, a, false, b, (short)0, acc, false, false);

        const float hov = ho[d0 + n];

        // Lanes 0-15: acc[0]=mixed row0, acc[1]=row1 (streams 0,1).
        // Lanes 16-31: acc[0]=mixed M=8 -> stream2, acc[1]=M=9 -> stream3.
        const float o0 = acc[0] * hov + hp0 * b.x;
        const float o1 = acc[1] * hov + hp1 * b.y;
        __builtin_nontemporal_store(o0, &ob[(j0    ) * D_DIM + d0 + n]);
        __builtin_nontemporal_store(o1, &ob[(j0 + 1) * D_DIM + d0 + n]);
    }
}

extern "C" void kernel_launch(void* const* d_in, const int* in_sizes, int n_in,
                              void* d_out, int out_size, void* d_ws, size_t ws_size,
                              hipStream_t stream) {
    (void)in_sizes; (void)n_in; (void)out_size; (void)d_ws; (void)ws_size;
    const float* x      = (const float*)d_in[0];
    const float* h_res  = (const float*)d_in[1];
    const float* h_out  = (const float*)d_in[2];
    const float* h_post = (const float*)d_in[3];
    float*       out    = (float*)d_out;

    // B*S*(D/256) = 65536 waves, 8 waves (256 threads) per block.
    const int total_waves = B_DIM * S_DIM * (D_DIM / 256);
    dim3 grid(total_waves / 8), block(256);
    stream_mix_wmma<<<grid, block, 0, stream>>>(x, h_res, h_out, h_post, out);
}